// WindowCorr_47760036331791
// MI455X (gfx1250) — compile-verified
//
#include <hip/hip_runtime.h>

// 3D sliding-window correlation (radius 1, 27 offsets) on MI455X / gfx1250.
// Per 16-wide w-tile: full 16x16 pairwise correlation tiles (X-tile vs Y-tile
// shifted by (dz,dy)) via V_WMMA_F32_16X16X4_F32, K-looped over 96 channels;
// the three w-diagonals of each C tile are the dx=-1/0/+1 outputs.

typedef __attribute__((ext_vector_type(2))) float v2f;
typedef __attribute__((ext_vector_type(8))) float v8f;

#define C_CH      96
#define D_D       64
#define H_H       128
#define W_W       128
#define CH_STRIDE (D_D * H_H * W_W)   // 1048576 floats (also out-plane stride)
#define ZSTR      (H_H * W_W)         // 16384
#define YSTR      (W_W)               // 128
#define TILE_OUT  14                  // complete outputs per 16-wide tile
#define NTILES    10                  // ceil(128/14)
#define SCALE     0.10206207261596577f // 96^-0.5

__global__ __launch_bounds__(256) void
WindowCorr_wmma_kernel(const float* __restrict__ x,
                       const float* __restrict__ y,
                       float* __restrict__ out) {
    const int lane  = threadIdx.x & 31;
    const int wave  = blockIdx.x * 8 + (threadIdx.x >> 5); // 8 waves / block
    // wave -> (z, y, w-tile)
    const int t  = wave % NTILES;
    const int zy = wave / NTILES;
    const int yy = zy % H_H;
    const int zz = zy / H_H;

    const int ws    = TILE_OUT * t - 1;   // tile covers w = ws .. ws+15
    const int nIdx  = lane & 15;          // M for A rows / N for B cols
    const int kHalf = lane >> 4;          // lane-half selects channel pair
    const int wcol  = ws + nIdx;
    const bool wvalid = (wcol >= 0) && (wcol < W_W);
    const int wclamp  = min(max(wcol, 0), W_W - 1);

    // Per-lane, loop-invariant i32 element offsets (< 2^25 elements). The
    // uniform pointer (px/py) is bumped per K-step -> saddr-form global loads.
    const int loffA = 2 * kHalf * CH_STRIDE + zz * ZSTR + yy * YSTR + wclamp;

    int   loffB[9];
    float maskB[9];
#pragma unroll
    for (int i = 0; i < 9; ++i) {
        const int dz = i / 3 - 1, dy = i % 3 - 1;
        const int zn = zz + dz, yn = yy + dy;
        const bool rowok = (zn >= 0) && (zn < D_D) && (yn >= 0) && (yn < H_H);
        const int znc = min(max(zn, 0), D_D - 1);
        const int ync = min(max(yn, 0), H_H - 1);
        loffB[i] = 2 * kHalf * CH_STRIDE + znc * ZSTR + ync * YSTR + wclamp;
        maskB[i] = (rowok && wvalid) ? 1.0f : 0.0f;   // zero-pad mask (B only)
    }

    v8f acc[9];
#pragma unroll
    for (int i = 0; i < 9; ++i) acc[i] = v8f{};

    // K loop: 4 channels per WMMA step; 2 A loads + 18 B loads + 9 WMMAs.
    // Uniform base pointers advance 16 MB/step; +CH_STRIDE folds into imm24.
    const float* px = x;
    const float* py = y;
    for (int k0 = 0; k0 < C_CH; k0 += 4) {
        v2f a;
        a.x = px[loffA] * SCALE;             // channel k0   (+2*kHalf)
        a.y = px[CH_STRIDE + loffA] * SCALE; // channel k0+1 (+2*kHalf)
#pragma unroll
        for (int i = 0; i < 9; ++i) {
            v2f b;
            b.x = py[loffB[i]] * maskB[i];
            b.y = py[CH_STRIDE + loffB[i]] * maskB[i];
            // D = A(16x4) * B(4x16) + C ; fp32 in/out, 8-arg VOP3P form
            acc[i] = __builtin_amdgcn_wmma_f32_16x16x4_f32(
                false, a, false, b, (short)0, acc[i], false, false);
        }
        px += (size_t)4 * CH_STRIDE;
        py += (size_t)4 * CH_STRIDE;
    }

    // Band extraction: C[m][n] lives in vgpr (m%8), lane (n + 16*(m>=8)).
    // Output: w = ws + m = 14*t + (m-1), dx index = n - m + 1 in {0,1,2}.
    // Output is write-only: nontemporal stores keep L2 free for y halo reuse.
    const int rowoff = zz * ZSTR + yy * YSTR;
#pragma unroll
    for (int r = 0; r < 8; ++r) {
        const int m    = r + 8 * kHalf;     // lane-half adds 8 to M
        const int j    = m - 1;             // output index within tile
        const int dxi  = nIdx - j;          // n - (m-1)
        const int wout = TILE_OUT * t + j;
        const bool ok = (j >= 0) && (j < TILE_OUT) &&
                        (dxi >= 0) && (dxi < 3) && (wout < W_W);
        if (ok) {
            const size_t base = (size_t)dxi * CH_STRIDE + rowoff + wout;
#pragma unroll
            for (int i = 0; i < 9; ++i) {   // o = i*3 + dxi, plane stride = 1M
                __builtin_nontemporal_store(acc[i][r],
                    &out[(size_t)(i * 3) * CH_STRIDE + base]);
            }
        }
    }
}

extern "C" void kernel_launch(void* const* d_in, const int* in_sizes, int n_in,
                              void* d_out, int out_size, void* d_ws, size_t ws_size,
                              hipStream_t stream) {
    const float* x = (const float*)d_in[0];
    const float* y = (const float*)d_in[1];
    float* out = (float*)d_out;
    // waves = 64*128*10 = 81920 ; 8 waves (256 threads) per block
    const int blocks = (D_D * H_H * NTILES) / 8; // 10240
    WindowCorr_wmma_kernel<<<blocks, 256, 0, stream>>>(x, y, out);
}